// HybridScoring_4226247819606
// MI455X (gfx1250) — compile-verified
//
#include <hip/hip_runtime.h>
#include <stdint.h>

// Problem constants (match reference).
constexpr int BATCH = 256;
constexpr int NP1   = 10000;
constexpr int KNN   = 32;
constexpr int BLOCK = 1024;              // 32 wave32s on one WGP
constexpr int NWAVE = BLOCK / 32;        // 32 waves
constexpr int PER   = (NP1 + BLOCK - 1) / BLOCK;  // 10 nodes per thread
#define NEG_INF_F (-1000000000.0f)

#define GAS __attribute__((address_space(1)))
#define LAS __attribute__((address_space(3)))
typedef int v4i __attribute__((ext_vector_type(4)));

__device__ __forceinline__ float wave_red_max(float v) {
#pragma unroll
  for (int o = 16; o > 0; o >>= 1) v = fmaxf(v, __shfl_down(v, o, 32));
  return v;
}
__device__ __forceinline__ float wave_red_sum(float v) {
#pragma unroll
  for (int o = 16; o > 0; o >>= 1) v += __shfl_down(v, o, 32);
  return v;
}

__global__ __launch_bounds__(BLOCK) void hybrid_scoring_kernel(
    const float*   __restrict__ query,   // [B,2]
    const float*   __restrict__ psi,     // [B,NP1,2]
    const int*     __restrict__ knn,     // [B,NP1,K]
    const uint8_t* __restrict__ mask,    // [B,NP1] (jnp bool = 1 byte/elem)
    const float*   __restrict__ cur,     // [B,2]
    const float*   __restrict__ coords,  // [B,NP1,2]
    const float*   __restrict__ lam_p,   // [1]
    const float*   __restrict__ mu_p,    // [1]
    float*         __restrict__ out)     // [B,NP1]
{
  // 80,000 B of psi + reduction scratch. WGP LDS is 320 KB, fits easily.
  __shared__ __align__(16) float s_psi[2 * NP1];
  __shared__ float s_red[NWAVE];
  __shared__ float s_bval;

  const int b    = blockIdx.x;
  const int tid  = threadIdx.x;
  const int lane = tid & 31;
  const int wid  = tid >> 5;

  const float lam = lam_p[0];
  const float mu  = mu_p[0];
  const float qx  = query[2 * b + 0], qy = query[2 * b + 1];
  const float cx  = cur[2 * b + 0],   cy = cur[2 * b + 1];

  // ---------- Phase 1: stage psi[b] (80 KB) into LDS via async DMA ----------
  const float* gsrc = psi + (size_t)b * (2 * NP1);
#if __has_builtin(__builtin_amdgcn_global_load_async_to_lds_b128)
  for (int c = tid; c < (2 * NP1) / 4; c += BLOCK) {     // 5000 x 16B chunks
    __builtin_amdgcn_global_load_async_to_lds_b128(
        (GAS v4i*)(gsrc + 4 * c), (LAS v4i*)(&s_psi[4 * c]),
        /*offset=*/0, /*cpol=*/0);
  }
#if __has_builtin(__builtin_amdgcn_s_wait_asynccnt)
  __builtin_amdgcn_s_wait_asynccnt(0);
#else
  asm volatile("s_wait_asynccnt 0" ::: "memory");
#endif
#else
  for (int c = tid; c < (2 * NP1) / 4; c += BLOCK) {
    ((float4*)s_psi)[c] = ((const float4*)gsrc)[c];
  }
#endif
  __syncthreads();

  // ---------- Phase 2: per-node scores (10 nodes/thread, kept in VGPRs) ----------
  const float2*  s2 = (const float2*)s_psi;                 // ds_load_b64 gathers
  const int*     kb = knn    + (size_t)b * NP1 * KNN;
  const float*   ab = coords + (size_t)b * NP1 * 2;
  const uint8_t* mb = mask   + (size_t)b * NP1;

  float sc[PER];
  float lmax = -INFINITY;

#pragma unroll
  for (int i = 0; i < PER; ++i) {
    const int n = tid + i * BLOCK;
    float v = -INFINITY;                 // padding lanes contribute exp() = 0
    if (n < NP1) {
      const float2 p = s2[n];            // conflict-free: lane-stride = 8B
      // sum_k psi[nb_k] first, then one dot: psi_n . (sum_k psi_nb)
      float sx = 0.f, sy = 0.f;
      const int4* kp = (const int4*)(kb + (size_t)n * KNN);
#pragma unroll
      for (int j = 0; j < KNN / 4; ++j) {   // 8 x global_load_b128, 128B/thread
        const int4 id = kp[j];
        const float2 n0 = s2[id.x];
        const float2 n1 = s2[id.y];
        const float2 n2 = s2[id.z];
        const float2 n3 = s2[id.w];
        sx += n0.x + n1.x + n2.x + n3.x;
        sy += n0.y + n1.y + n2.y + n3.y;
      }
      const float ctx   = p.x * qx + p.y * qy;
      const float inter = p.x * sx + p.y * sy;
      const float2 ac = ((const float2*)ab)[n];
      const float dx = ac.x - cx, dy = ac.y - cy;
      const float dist = sqrtf(dx * dx + dy * dy);
      v = ctx + lam * inter - mu * dist;
      if (mb[n]) v = NEG_INF_F;
    }
    sc[i] = v;
    lmax = fmaxf(lmax, v);
  }

  // ---------- Phase 3: block-wide log-softmax ----------
  // block max
  float w = wave_red_max(lmax);
  if (lane == 0) s_red[wid] = w;
  __syncthreads();
  if (wid == 0) {
    float t = (lane < NWAVE) ? s_red[lane] : -INFINITY;
    t = wave_red_max(t);
    if (lane == 0) s_bval = t;
  }
  __syncthreads();
  const float gmax = s_bval;

  // block sum of exp(x - max)
  float lsum = 0.f;
#pragma unroll
  for (int i = 0; i < PER; ++i) lsum += expf(sc[i] - gmax);

  float ws = wave_red_sum(lsum);
  if (lane == 0) s_red[wid] = ws;
  __syncthreads();
  if (wid == 0) {
    float t = (lane < NWAVE) ? s_red[lane] : 0.f;
    t = wave_red_sum(t);
    if (lane == 0) s_bval = t;
  }
  __syncthreads();
  const float gl = gmax + logf(s_bval);   // logsumexp

  // ---------- write out: coalesced b32 stores ----------
#pragma unroll
  for (int i = 0; i < PER; ++i) {
    const int n = tid + i * BLOCK;
    if (n < NP1) out[(size_t)b * NP1 + n] = sc[i] - gl;
  }
}

extern "C" void kernel_launch(void* const* d_in, const int* in_sizes, int n_in,
                              void* d_out, int out_size, void* d_ws, size_t ws_size,
                              hipStream_t stream) {
  (void)in_sizes; (void)n_in; (void)out_size; (void)d_ws; (void)ws_size;
  const float*   query  = (const float*)d_in[0];
  const float*   psi    = (const float*)d_in[1];
  const int*     knn    = (const int*)d_in[2];
  const uint8_t* mask   = (const uint8_t*)d_in[3];
  const float*   cur    = (const float*)d_in[4];
  const float*   coords = (const float*)d_in[5];
  const float*   lam    = (const float*)d_in[6];
  const float*   mu     = (const float*)d_in[7];

  hybrid_scoring_kernel<<<BATCH, BLOCK, 0, stream>>>(
      query, psi, knn, mask, cur, coords, lam, mu, (float*)d_out);
}